// FlatLogSoftmaxNLL_37555194036883
// MI455X (gfx1250) — compile-verified
//
#include <hip/hip_runtime.h>
#include <hip/hip_bf16.h>
#include <math.h>

#define LEAVES 10000
#define NCHUNK (LEAVES / 4)   // 2500 float4 / uchar4 chunks, no tail
#define NTHREADS 256
#define NWAVES (NTHREADS / 32)

// 16-byte GCC-style vector type matching the builtin's parameter pointee type
typedef int v4i __attribute__((vector_size(16)));

// ---------------------------------------------------------------------------
// CDNA5 async global->LDS staging (ASYNCcnt path). Confirmed lowering to
// global_load_async_to_lds_b128 in round 2.
// ---------------------------------------------------------------------------
__device__ __forceinline__ void async_copy_b128(const void* gsrc, void* ldst) {
#if __has_builtin(__builtin_amdgcn_global_load_async_to_lds_b128)
  __builtin_amdgcn_global_load_async_to_lds_b128(
      (__attribute__((address_space(1))) v4i*)gsrc,
      (__attribute__((address_space(3))) v4i*)ldst,
      /*imm offset*/ 0, /*cpol*/ 0);
#else
  *(float4*)ldst = *(const float4*)gsrc;  // synchronous fallback
#endif
}

__device__ __forceinline__ void wait_async0() {
#if __has_builtin(__builtin_amdgcn_s_wait_asynccnt)
  __builtin_amdgcn_s_wait_asynccnt(0);
#elif __has_builtin(__builtin_amdgcn_global_load_async_to_lds_b128)
  asm volatile("s_wait_asynccnt 0" ::: "memory");
#endif
}

// ---------------------------------------------------------------------------
// Kernel 1: one workgroup (8 x wave32) per batch row.
//   row_logp[b] = log(sum_masked exp(s - m)) - log(sum_all exp(s - m))
// (the shared max m cancels in the difference; used only for exp range)
// ---------------------------------------------------------------------------
__global__ __launch_bounds__(NTHREADS) void hs_row_kernel(
    const float* __restrict__ scores,        // [B, LEAVES]
    const int* __restrict__ labels,          // [B]
    const unsigned char* __restrict__ anc,   // [NODES, LEAVES] bool (0/1 bytes)
    float* __restrict__ row_logp)            // [B]
{
  __shared__ __align__(16) float         s_scores[LEAVES];
  __shared__ __align__(16) unsigned char s_mask[LEAVES];
  __shared__ float s_redA[NWAVES];
  __shared__ float s_redB[NWAVES];

  const int row  = blockIdx.x;
  const int tid  = threadIdx.x;
  const int lane = tid & 31;
  const int wid  = tid >> 5;

  const size_t srow = (size_t)row * LEAVES;
  const int    label = labels[row];
  const size_t mrow = (size_t)label * LEAVES;   // 10000 % 16 == 0 -> 16B aligned

  // Stage score row: 2500 x 16B async chunks
  for (int i = tid; i < NCHUNK; i += NTHREADS)
    async_copy_b128(scores + srow + (size_t)i * 4, s_scores + i * 4);
  // Stage gathered mask row: 625 x 16B async chunks
  for (int i = tid; i < LEAVES / 16; i += NTHREADS)
    async_copy_b128(anc + mrow + (size_t)i * 16, s_mask + i * 16);

  wait_async0();
  __syncthreads();

  // ---- pass 1: row max, vectorized ds_load_b128 sweep ----
  float m = -INFINITY;
  for (int c = tid; c < NCHUNK; c += NTHREADS) {
    const float4 v = *(const float4*)(s_scores + 4 * c);
    m = fmaxf(m, fmaxf(fmaxf(v.x, v.y), fmaxf(v.z, v.w)));
  }
#pragma unroll
  for (int off = 16; off >= 1; off >>= 1)
    m = fmaxf(m, __shfl_xor(m, off, 32));
  if (lane == 0) s_redA[wid] = m;
  __syncthreads();
  float mb = s_redA[0];
#pragma unroll
  for (int w = 1; w < NWAVES; ++w) mb = fmaxf(mb, s_redA[w]);
  __syncthreads();  // protect s_redA before reuse

  // ---- pass 2: full + masked exp-sums, b128 scores + b32 mask per chunk ----
  float sf = 0.f, sm = 0.f;
  for (int c = tid; c < NCHUNK; c += NTHREADS) {
    const float4   v  = *(const float4*)(s_scores + 4 * c);
    const unsigned mk = *(const unsigned*)(s_mask + 4 * c);  // 4 bool bytes
    const float e0 = __expf(v.x - mb);
    const float e1 = __expf(v.y - mb);
    const float e2 = __expf(v.z - mb);
    const float e3 = __expf(v.w - mb);
    sf += (e0 + e1) + (e2 + e3);
    // bool bytes are exactly 0 or 1: use them as multiplicands
    float me = (float)(mk & 0xFFu) * e0;
    me = fmaf((float)((mk >> 8) & 0xFFu), e1, me);
    me = fmaf((float)((mk >> 16) & 0xFFu), e2, me);
    me = fmaf((float)((mk >> 24) & 0xFFu), e3, me);
    sm += me;
  }
#pragma unroll
  for (int off = 16; off >= 1; off >>= 1) {
    sf += __shfl_xor(sf, off, 32);
    sm += __shfl_xor(sm, off, 32);
  }
  if (lane == 0) { s_redA[wid] = sf; s_redB[wid] = sm; }
  __syncthreads();

  if (tid == 0) {
    float tf = 0.f, tm = 0.f;
#pragma unroll
    for (int w = 0; w < NWAVES; ++w) { tf += s_redA[w]; tm += s_redB[w]; }
    row_logp[row] = logf(tm) - logf(tf);
  }
}

// ---------------------------------------------------------------------------
// Kernel 2: deterministic single-block mean reduction, out = -mean(row_logp)
// ---------------------------------------------------------------------------
__global__ __launch_bounds__(NTHREADS) void hs_reduce_kernel(
    const float* __restrict__ row_logp, float* __restrict__ out, int n)
{
  __shared__ float s_red[NWAVES];
  const int tid  = threadIdx.x;
  const int lane = tid & 31;
  const int wid  = tid >> 5;

  float s = 0.f;
  for (int i = tid; i < n; i += NTHREADS) s += row_logp[i];
#pragma unroll
  for (int off = 16; off >= 1; off >>= 1) s += __shfl_xor(s, off, 32);
  if (lane == 0) s_red[wid] = s;
  __syncthreads();
  if (tid == 0) {
    float t = 0.f;
#pragma unroll
    for (int w = 0; w < NWAVES; ++w) t += s_red[w];
    out[0] = -(t / (float)n);
  }
}

// ---------------------------------------------------------------------------
// Harness entry point
// ---------------------------------------------------------------------------
extern "C" void kernel_launch(void* const* d_in, const int* in_sizes, int n_in,
                              void* d_out, int out_size, void* d_ws, size_t ws_size,
                              hipStream_t stream) {
  const float*         scores = (const float*)d_in[0];
  const int*           labels = (const int*)d_in[1];
  const unsigned char* anc    = (const unsigned char*)d_in[2];
  float* out = (float*)d_out;
  float* ws  = (float*)d_ws;   // B floats of per-row logp

  const int B = in_sizes[1];   // 4096 labels

  hs_row_kernel<<<B, NTHREADS, 0, stream>>>(scores, labels, anc, ws);
  hs_reduce_kernel<<<1, NTHREADS, 0, stream>>>(ws, out, B);
}